// GCN_29867202577012
// MI455X (gfx1250) — compile-verified
//
#include <hip/hip_runtime.h>
#include <hip/hip_bf16.h>

// ---------------------------------------------------------------------------
// GCN (3x GCNConv H=128) + readout MLP for MI455X (gfx1250, wave32, WMMA).
//
// - GEMMs: V_WMMA_F32_16X16X4_F32, one wave per 16x128 output stripe,
//   A staged via conflict-free LDS (ds_load_b64 fragments), B pre-packed as
//   K-pairs so every fragment is one coalesced global_load_b64 with an
//   immediate offset (K-loop fully unrolled -> no per-iter address math).
// - Embedding gather fused into GEMM-1 staging; ReLU fused into GEMM-2/3
//   input staging (saves three full 100k x 128 f32 passes).
// - Sparse aggregation: wave32 per edge, float4 per lane (512B coalesced),
//   global_atomic_add_f32 scatter. Feature matrices (51.2MB) are L2-resident
//   in the 192MB L2, so the 5.2GB of irregular gather/atomic traffic runs at
//   L2, not HBM, speed.
// ---------------------------------------------------------------------------

typedef float v2f __attribute__((ext_vector_type(2)));
typedef float v8f __attribute__((ext_vector_type(8)));

#define HDIM 128
#define H4   32           // HDIM/4
#define LROW 132          // padded LDS row stride (floats): 132%64=4 -> no bank conflicts

// ---------------- degree / normalization ----------------

__global__ __launch_bounds__(256) void k_init_deg(float* __restrict__ deg, int n) {
    int i = blockIdx.x * blockDim.x + threadIdx.x;
    if (i < n) deg[i] = 1.0f;                       // self-loop
}

__global__ __launch_bounds__(256) void k_deg_atomic(const int* __restrict__ ei,
                                                    float* __restrict__ deg, int E) {
    int e = blockIdx.x * blockDim.x + threadIdx.x;
    if (e < E) atomicAdd(&deg[ei[E + e]], 1.0f);
}

__global__ __launch_bounds__(256) void k_dinv(const float* __restrict__ deg,
                                              float* __restrict__ dinv, int n) {
    int i = blockIdx.x * blockDim.x + threadIdx.x;
    if (i < n) dinv[i] = rsqrtf(deg[i]);            // deg >= 1 always
}

// ---------------- weight pre-pack: Wp[k/2][col] = {W[k][col], W[k+1][col]} --

__global__ __launch_bounds__(256) void k_pack_w(const float* __restrict__ W,
                                                float2* __restrict__ Wp) {
    int t = blockIdx.x * blockDim.x + threadIdx.x;  // 64*128 threads
    if (t >= 64 * HDIM) return;
    int k2  = t >> 7;
    int col = t & 127;
    float2 v;
    v.x = W[(size_t)(2 * k2 + 0) * HDIM + col];
    v.y = W[(size_t)(2 * k2 + 1) * HDIM + col];
    Wp[t] = v;
}

// ---------------- WMMA f32 GEMM: Y[M,128] = X[M,128] @ W[128,128] ----------
// Requires M % 16 == 0. One wave per 16-row stripe; 8 f32 16x16x4 accumulators.
// A: wave stages its own 16x128 stripe into LDS (coalesced b128 loads, padded
//    rows), fragments via ds_load_b64 (lanes 0-15: K={k,k+1}, 16-31: K={k+2,k+3}).
// B: packed K-pair layout -> one global_load_b64 per fragment, imm offsets.
// Options: zidx (gather rows through z -> embedding fusion), reluIn, bias+reluOut.

__global__ __launch_bounds__(256) void k_gemm_wmma(const float* __restrict__ X,
                                                   const int* __restrict__ zidx,
                                                   const float2* __restrict__ Wp,
                                                   const float* __restrict__ bias,
                                                   float* __restrict__ Y,
                                                   int M, int reluIn, int reluOut) {
    int wid  = (blockIdx.x * blockDim.x + threadIdx.x) >> 5;   // global wave id
    int row0 = wid * 16;
    if (row0 >= M) return;                                     // uniform exit
    int lane = threadIdx.x & 31;
    int half = lane >> 4;
    int mrow = lane & 15;
    int wv   = threadIdx.x >> 5;                               // wave in block

    __shared__ float xs[8][16 * LROW];                         // 67.6 KB, per-wave slices
    float* xw = xs[wv];

    // ---- stage A stripe (coalesced: one full 512B row per iteration) ----
    const float4* X4 = (const float4*)X;
    for (int i = 0; i < 16; ++i) {
        int r = row0 + i;
        size_t srow = zidx ? (size_t)zidx[r] : (size_t)r;      // embedding fusion
        float4 v = X4[srow * H4 + lane];
        if (reluIn) {
            v.x = fmaxf(v.x, 0.f); v.y = fmaxf(v.y, 0.f);
            v.z = fmaxf(v.z, 0.f); v.w = fmaxf(v.w, 0.f);
        }
        *(float4*)&xw[i * LROW + lane * 4] = v;
    }
    // LDS ops are in-order within a wave and each wave reads only its own
    // stripe -> no block barrier needed.

    v8f acc[8] = {};
    const float2* wp = Wp + (size_t)half * HDIM + mrow;        // + t*16 + k/2*128
    const float*  ap = &xw[mrow * LROW + half * 2];            // + k

#pragma unroll
    for (int k = 0; k < HDIM; k += 4) {
        v2f a = *(const v2f*)(ap + k);                         // ds_load_b64
#pragma unroll
        for (int t = 0; t < 8; ++t) {
            float2 w = wp[(size_t)(k >> 1) * HDIM + t * 16];   // global_load_b64
            v2f b; b.x = w.x; b.y = w.y;
            acc[t] = __builtin_amdgcn_wmma_f32_16x16x4_f32(
                false, a, false, b, (short)0, acc[t], false, false);
        }
    }

    // ---- store: VGPR j -> rows {row0+j, row0+j+8} for lane halves ----
#pragma unroll
    for (int t = 0; t < 8; ++t) {
        int col = t * 16 + mrow;
        float bv = bias ? bias[col] : 0.0f;
#pragma unroll
        for (int j = 0; j < 8; ++j) {
            int r = row0 + j + half * 8;
            float v = acc[t][j] + bv;
            if (reluOut) v = fmaxf(v, 0.0f);
            Y[(size_t)r * HDIM + col] = v;
        }
    }
}

// ---------------- out init + edge scatter ----------------

__global__ __launch_bounds__(256) void k_init_bias(float* __restrict__ out,
                                                   const float* __restrict__ bias, int n) {
    int t = blockIdx.x * blockDim.x + threadIdx.x;
    if (t >= n * H4) return;
    ((float4*)out)[t] = ((const float4*)bias)[t & 31];
}

__global__ __launch_bounds__(256) void k_scatter(const float* __restrict__ h,
                                                 const float* __restrict__ dinv,
                                                 const int* __restrict__ ei,
                                                 int E, int N,
                                                 float* __restrict__ out) {
    int wid = (blockIdx.x * blockDim.x + threadIdx.x) >> 5;
    if (wid >= E + N) return;
    int lane = threadIdx.x & 31;
    if (lane == 0 && wid + 4096 < E) {                 // stream-ahead prefetch
        __builtin_prefetch(&ei[wid + 4096], 0, 0);
        __builtin_prefetch(&ei[E + wid + 4096], 0, 0);
    }
    int s, d;
    if (wid < E) { s = ei[wid]; d = ei[E + wid]; }
    else         { s = d = wid - E; }
    float nrm = dinv[s] * dinv[d];
    float4 v = ((const float4*)(h + (size_t)s * HDIM))[lane];
    v.x *= nrm; v.y *= nrm; v.z *= nrm; v.w *= nrm;
    float* od = out + (size_t)d * HDIM + lane * 4;
    atomicAdd(od + 0, v.x);
    atomicAdd(od + 1, v.y);
    atomicAdd(od + 2, v.z);
    atomicAdd(od + 3, v.w);
}

// ---------------- readout ----------------

__global__ __launch_bounds__(256) void k_centers(const int* __restrict__ batch,
                                                 int* __restrict__ centers,
                                                 int N, int G) {
    int g = blockIdx.x * blockDim.x + threadIdx.x;
    if (g >= G) return;
    int lo = 0, hi = N;                                // lower_bound(batch, g)
    while (lo < hi) { int mid = (lo + hi) >> 1; if (batch[mid] < g) lo = mid + 1; else hi = mid; }
    centers[g] = lo;
}

// p[g] = x[center]*x[center+1]; rows G..Gpad zero-filled so the readout GEMM
// needs no bounds checks.
__global__ __launch_bounds__(256) void k_pair(const float* __restrict__ x,
                                              const int* __restrict__ centers,
                                              float* __restrict__ p, int G, int Gpad) {
    int t = blockIdx.x * blockDim.x + threadIdx.x;
    if (t >= Gpad * H4) return;
    int g  = t >> 5;
    int c4 = t & 31;
    float4 r = {0.f, 0.f, 0.f, 0.f};
    if (g < G) {
        int c = centers[g];
        float4 a = ((const float4*)x)[(size_t)c * H4 + c4];
        float4 b = ((const float4*)x)[(size_t)(c + 1) * H4 + c4];
        r.x = a.x * b.x; r.y = a.y * b.y; r.z = a.z * b.z; r.w = a.w * b.w;
    }
    ((float4*)p)[t] = r;
}

__global__ __launch_bounds__(256) void k_final(const float* __restrict__ hbuf,
                                               const float* __restrict__ mw2,
                                               const float* __restrict__ mb2,
                                               float* __restrict__ out, int G) {
    int g = blockIdx.x * blockDim.x + threadIdx.x;
    if (g >= G) return;
    float acc = mb2[0];
    const float* hr = hbuf + (size_t)g * HDIM;
#pragma unroll 4
    for (int c = 0; c < HDIM; ++c) acc = fmaf(hr[c], mw2[c], acc);
    out[g] = acc;
}

// ---------------- host orchestration ----------------

static inline int cdiv(long long a, long long b) { return (int)((a + b - 1) / b); }

extern "C" void kernel_launch(void* const* d_in, const int* in_sizes, int n_in,
                              void* d_out, int out_size, void* d_ws, size_t ws_size,
                              hipStream_t stream) {
    const int N    = in_sizes[1];
    const int E    = in_sizes[2] / 2;
    const int G    = out_size;
    const int Gpad = (G + 15) & ~15;

    const int*   z     = (const int*)  d_in[1];
    const int*   ei    = (const int*)  d_in[2];
    const int*   batch = (const int*)  d_in[3];
    const float* ztab  = (const float*)d_in[5];
    const float* W1 = (const float*)d_in[6],  *b1 = (const float*)d_in[7];
    const float* W2 = (const float*)d_in[8],  *b2 = (const float*)d_in[9];
    const float* W3 = (const float*)d_in[10], *b3 = (const float*)d_in[11];
    const float* mw1 = (const float*)d_in[12], *mb1 = (const float*)d_in[13];
    const float* mw2 = (const float*)d_in[14], *mb2 = (const float*)d_in[15];
    float* out = (float*)d_out;

    // workspace carve-out (256B aligned slices)
    char*  ws  = (char*)d_ws;
    size_t off = 0;
    auto carve = [&](size_t bytes) -> void* {
        void* p = ws + off;
        off = (off + bytes + 255) & ~(size_t)255;
        return p;
    };
    float*  deg     = (float*) carve((size_t)N * 4);
    float*  dinv    = (float*) carve((size_t)N * 4);
    float*  bufA    = (float*) carve((size_t)N * HDIM * 4);
    float*  bufB    = (float*) carve((size_t)N * HDIM * 4);
    float*  bufC    = (float*) carve((size_t)N * HDIM * 4);
    float*  pbuf    = (float*) carve((size_t)Gpad * HDIM * 4);
    float*  hbuf    = (float*) carve((size_t)Gpad * HDIM * 4);
    int*    centers = (int*)   carve((size_t)G * 4);
    float2* Wp1  = (float2*)carve((size_t)64 * HDIM * 8);
    float2* Wp2  = (float2*)carve((size_t)64 * HDIM * 8);
    float2* Wp3  = (float2*)carve((size_t)64 * HDIM * 8);
    float2* Wpm  = (float2*)carve((size_t)64 * HDIM * 8);
    (void)ws_size; (void)n_in;

    const int TB = 256;
    const int nodeBlocks     = cdiv(N, TB);
    const int featBlocks     = cdiv((long long)N * H4, TB);
    const int edgeWaveBlocks = cdiv((long long)(E + N) * 32, TB);
    const int packBlocks     = cdiv(64 * HDIM, TB);
    const int gemmBlocksN    = cdiv((long long)(N / 16) * 32, TB);
    const int gemmBlocksG    = cdiv((long long)(Gpad / 16) * 32, TB);

    // normalization
    k_init_deg  <<<nodeBlocks, TB, 0, stream>>>(deg, N);
    k_deg_atomic<<<cdiv(E, TB), TB, 0, stream>>>(ei, deg, E);
    k_dinv      <<<nodeBlocks, TB, 0, stream>>>(deg, dinv, N);

    // weight pre-pack (K-pair layout for b64 B-fragment loads)
    k_pack_w<<<packBlocks, TB, 0, stream>>>(W1,  Wp1);
    k_pack_w<<<packBlocks, TB, 0, stream>>>(W2,  Wp2);
    k_pack_w<<<packBlocks, TB, 0, stream>>>(W3,  Wp3);
    k_pack_w<<<packBlocks, TB, 0, stream>>>(mw1, Wpm);

    // layer 1 (embedding gather fused into GEMM staging): ztab[z] @ W1 -> bufB
    k_gemm_wmma<<<gemmBlocksN, TB, 0, stream>>>(ztab, z, Wp1, nullptr, bufB, N, 0, 0);
    k_init_bias<<<featBlocks, TB, 0, stream>>>(bufC, b1, N);
    k_scatter  <<<edgeWaveBlocks, TB, 0, stream>>>(bufB, dinv, ei, E, N, bufC);

    // layer 2 (ReLU fused into input staging): relu(bufC) @ W2 -> bufB
    k_gemm_wmma<<<gemmBlocksN, TB, 0, stream>>>(bufC, nullptr, Wp2, nullptr, bufB, N, 1, 0);
    k_init_bias<<<featBlocks, TB, 0, stream>>>(bufA, b2, N);
    k_scatter  <<<edgeWaveBlocks, TB, 0, stream>>>(bufB, dinv, ei, E, N, bufA);

    // layer 3: relu(bufA) @ W3 -> bufB ; aggregate -> bufC (no output relu)
    k_gemm_wmma<<<gemmBlocksN, TB, 0, stream>>>(bufA, nullptr, Wp3, nullptr, bufB, N, 1, 0);
    k_init_bias<<<featBlocks, TB, 0, stream>>>(bufC, b3, N);
    k_scatter  <<<edgeWaveBlocks, TB, 0, stream>>>(bufB, dinv, ei, E, N, bufC);

    // readout
    k_centers<<<cdiv(G, TB), TB, 0, stream>>>(batch, centers, N, G);
    k_pair   <<<cdiv((long long)Gpad * H4, TB), TB, 0, stream>>>(bufC, centers, pbuf, G, Gpad);
    k_gemm_wmma<<<gemmBlocksG, TB, 0, stream>>>(pbuf, nullptr, Wpm, mb1, hbuf, Gpad, 0, 1);
    k_final  <<<cdiv(G, TB), TB, 0, stream>>>(hbuf, mw2, mb2, out, G);
}